// ViterbiDecoder_16956530885119
// MI455X (gfx1250) — compile-verified
//
#include <hip/hip_runtime.h>

#define NEGV  (-1000000000.0f)
#define BDIM  128
#define NROW  128
#define MCOL  128

__device__ __forceinline__ float lse2(float a, float b){
  float m = fmaxf(a, b);
  return m + __logf(__expf(a - m) + __expf(b - m));
}
__device__ __forceinline__ float lse3(float a, float b, float c){
  float m = fmaxf(fmaxf(a, b), c);
  return m + __logf(__expf(a - m) + __expf(b - m) + __expf(c - m));
}
// wave32 lane gather (bits [6:2] of byte index select the lane on gfx1250)
__device__ __forceinline__ float bperm(float x, int srclane){
  int idx = (srclane < 0 ? 0 : srclane) << 2;
  return __int_as_float(__builtin_amdgcn_ds_bpermute(idx, __float_as_int(x)));
}
// CDNA5 async copy: 12 bytes global -> LDS, tracked on ASYNCcnt.
// inst_offset is added to BOTH the LDS and global address (ISA ch.10 async ops).
__device__ __forceinline__ void async_row12(unsigned ldsoff, const float* g){
  asm volatile(
    "global_load_async_to_lds_b32 %0, %1, off\n\t"
    "global_load_async_to_lds_b32 %0, %1, off offset:4\n\t"
    "global_load_async_to_lds_b32 %0, %1, off offset:8"
    :: "v"(ldsoff), "v"(g) : "memory");
}
#define WAIT_ASYNC(n) asm volatile("s_wait_asynccnt %0" :: "i"(n) : "memory")
// low 32 bits of a flat pointer into the LDS aperture == group-segment offset
__device__ __forceinline__ unsigned ldsa(const void* p){
  return (unsigned)(size_t)p;
}
// async loads retire in order: <=6 outstanding proves current row landed while
// two more rows stream; tighten at the tail.
#define WAIT_ROW(i)                                   \
  do {                                                \
    if ((i) + 2 <= NROW - 1)      { WAIT_ASYNC(6); }  \
    else if ((i) + 1 <= NROW - 1) { WAIT_ASYNC(3); }  \
    else                          { WAIT_ASYNC(0); }  \
  } while (0)

// ---------------- forward: alpha -> out buffer, logZ -> ws ----------------
__global__ __launch_bounds__(BDIM) void fb_forward(
    const float* __restrict__ theta, const float* __restrict__ A,
    float* __restrict__ alpha, float* __restrict__ logZ)
{
  const int b    = blockIdx.x;
  const int j    = threadIdx.x;
  const int lane = j & 31;
  const int wid  = j >> 5;

  __shared__ float sPrev[2][MCOL][3];
  __shared__ float sMX[MCOL][2];
  __shared__ float sAggU[4], sAggV[4];
  __shared__ float sTh[4][MCOL * 3];     // 4-deep prefetch ring

  float Ar[9];
  const float* Ab = A + b * 9;
  #pragma unroll
  for (int k = 0; k < 9; ++k) Ar[k] = Ab[k];   // Ar[k*3+s] = A[k][s]

  const float* thB = theta + (size_t)b * NROW * MCOL * 3;
  float*       alB = alpha + (size_t)b * NROW * MCOL * 3;

  sPrev[0][j][0] = NEGV; sPrev[0][j][1] = NEGV; sPrev[0][j][2] = NEGV;

  #pragma unroll
  for (int r = 0; r < 3; ++r)
    async_row12(ldsa(&sTh[r][j * 3]), thB + (size_t)(r * MCOL + j) * 3);
  __syncthreads();

  float mL = NEGV, xL = NEGV, yL = NEGV;       // last-row cell (for logZ)

  for (int i = 0; i < NROW; ++i){
    const int cur = i & 1;
    const int nxt = cur ^ 1;
    const int tb  = i & 3;

    WAIT_ROW(i);
    if (i + 3 < NROW)
      async_row12(ldsa(&sTh[(i + 3) & 3][j * 3]), thB + (size_t)((i + 3) * MCOL + j) * 3);

    const float th0 = sTh[tb][j * 3 + 0];
    const float th1 = sTh[tb][j * 3 + 1];
    const float th2 = sTh[tb][j * 3 + 2];

    const float pm = sPrev[cur][j][0], px = sPrev[cur][j][1], py = sPrev[cur][j][2];
    float dm = NEGV, dx = NEGV, dy = NEGV;
    if (j > 0){ dm = sPrev[cur][j-1][0]; dx = sPrev[cur][j-1][1]; dy = sPrev[cur][j-1][2]; }

    float lm = lse3(dm + Ar[0], dx + Ar[3], dy + Ar[6]);   // LSE_k(diag[k] + A[k][0])
    if (i == 0 && j == 0) lm = 0.0f;
    const float lx = lse3(pm + Ar[1], px + Ar[4], py + Ar[7]);
    const float m  = th0 + lm;
    const float x  = th1 + lx;

    sMX[j][0] = m; sMX[j][1] = x;
    __syncthreads();

    const float mp = (j > 0) ? sMX[j-1][0] : NEGV;
    const float xp = (j > 0) ? sMX[j-1][1] : NEGV;
    // y_j = LSE(U_j, V_j + y_{j-1})
    float U = th2 + lse2(mp + Ar[2], xp + Ar[5]);
    float V = th2 + Ar[8];

    #pragma unroll
    for (int off = 1; off < 32; off <<= 1){
      float Up = bperm(U, lane - off);
      float Vp = bperm(V, lane - off);
      if (lane >= off){ U = lse2(U, V + Up); V = V + Vp; }
    }
    if (lane == 31){ sAggU[wid] = U; sAggV[wid] = V; }
    __syncthreads();

    float ybase = NEGV;                 // y_{-1}
    #pragma unroll
    for (int w = 0; w < 3; ++w)
      if (w < wid) ybase = lse2(sAggU[w], sAggV[w] + ybase);
    const float y = lse2(U, V + ybase);

    sPrev[nxt][j][0] = m; sPrev[nxt][j][1] = x; sPrev[nxt][j][2] = y;
    float* ao = alB + (size_t)(i * MCOL + j) * 3;
    ao[0] = m; ao[1] = x; ao[2] = y;
    mL = m; xL = x; yL = y;
    __syncthreads();
  }
  if (j == MCOL - 1) logZ[b] = lse3(mL, xL, yL);
}

// --------- backward on flipped grid, fused combine: out = alpha + beta - logZ ---------
__global__ __launch_bounds__(BDIM) void fb_backward(
    const float* __restrict__ theta, const float* __restrict__ A,
    const float* __restrict__ logZ, float* __restrict__ out)
{
  const int b    = blockIdx.x;
  const int jf   = threadIdx.x;
  const int lane = jf & 31;
  const int wid  = jf >> 5;

  __shared__ float sU[2][MCOL][3];     // u = g + theta_f rows
  __shared__ float sC[MCOL];           // carry u[2] for left-shift
  __shared__ float sAggU[4], sAggV[4];
  __shared__ float sTh[4][MCOL * 3];

  float Ar[9];
  const float* Ab = A + b * 9;
  #pragma unroll
  for (int k = 0; k < 9; ++k) Ar[k] = Ab[k];
  const float lz = logZ[b];

  const float* thB = theta + (size_t)b * NROW * MCOL * 3;
  float*       oB  = out   + (size_t)b * NROW * MCOL * 3;

  sU[0][jf][0] = NEGV; sU[0][jf][1] = NEGV; sU[0][jf][2] = NEGV;

  // theta_f[p, jf, s] = theta[N-1-p, M-1-jf, s]
  #pragma unroll
  for (int r = 0; r < 3; ++r)
    async_row12(ldsa(&sTh[r][jf * 3]),
                thB + (size_t)((NROW - 1 - r) * MCOL + (MCOL - 1 - jf)) * 3);
  __syncthreads();

  for (int p = 0; p < NROW; ++p){
    const int cur = p & 1;
    const int nxt = cur ^ 1;
    const int tb  = p & 3;

    WAIT_ROW(p);
    if (p + 3 < NROW)
      async_row12(ldsa(&sTh[(p + 3) & 3][jf * 3]),
                  thB + (size_t)((NROW - 4 - p) * MCOL + (MCOL - 1 - jf)) * 3);

    const float th0 = sTh[tb][jf * 3 + 0];
    const float th1 = sTh[tb][jf * 3 + 1];
    const float th2 = sTh[tb][jf * 3 + 2];

    const float tm = (jf > 0) ? sU[cur][jf-1][0] : NEGV;   // u_prev[jf-1][0]
    const float tx = sU[cur][jf][1];                        // u_prev[jf][1]

    // carry c_j = th2 + LSE(A[2][0]+tm, A[2][1]+tx, A[2][2]+c_{j-1})
    float U = th2 + lse2(Ar[6] + tm, Ar[7] + tx);
    float V = th2 + Ar[8];
    if (p == 0 && jf == 0){ U = th2; V = NEGV; }           // g==0 at flipped origin

    #pragma unroll
    for (int off = 1; off < 32; off <<= 1){
      float Up = bperm(U, lane - off);
      float Vp = bperm(V, lane - off);
      if (lane >= off){ U = lse2(U, V + Up); V = V + Vp; }
    }
    if (lane == 31){ sAggU[wid] = U; sAggV[wid] = V; }
    __syncthreads();

    float ybase = NEGV;
    #pragma unroll
    for (int w = 0; w < 3; ++w)
      if (w < wid) ybase = lse2(sAggU[w], sAggV[w] + ybase);
    const float c = lse2(U, V + ybase);

    sC[jf] = c;
    __syncthreads();
    const float cp = (jf > 0) ? sC[jf-1] : NEGV;

    float g0 = lse3(Ar[0] + tm, Ar[1] + tx, Ar[2] + cp);   // LSE_k(A[0][k] + v[k])
    float g1 = lse3(Ar[3] + tm, Ar[4] + tx, Ar[5] + cp);
    float g2 = lse3(Ar[6] + tm, Ar[7] + tx, Ar[8] + cp);
    if (p == 0 && jf == 0){ g0 = 0.0f; g1 = 0.0f; g2 = 0.0f; }

    sU[nxt][jf][0] = g0 + th0;
    sU[nxt][jf][1] = g1 + th1;
    sU[nxt][jf][2] = c;                                    // == g2 + th2

    float* oo = oB + (size_t)((NROW - 1 - p) * MCOL + (MCOL - 1 - jf)) * 3;
    oo[0] = oo[0] + g0 - lz;                               // oo[] holds alpha
    oo[1] = oo[1] + g1 - lz;
    oo[2] = oo[2] + g2 - lz;
    __syncthreads();
  }
}

extern "C" void kernel_launch(void* const* d_in, const int* in_sizes, int n_in,
                              void* d_out, int out_size, void* d_ws, size_t ws_size,
                              hipStream_t stream)
{
  const float* theta = (const float*)d_in[0];
  const float* A     = (const float*)d_in[1];
  float* out  = (float*)d_out;
  float* logZ = (float*)d_ws;            // B floats of scratch
  const int B = in_sizes[1] / 9;

  fb_forward <<<dim3(B), dim3(BDIM), 0, stream>>>(theta, A, out, logZ);
  fb_backward<<<dim3(B), dim3(BDIM), 0, stream>>>(theta, A, logZ, out);
}